// DySample_20023137534392
// MI455X (gfx1250) — compile-verified
//
#include <hip/hip_runtime.h>

typedef float v2f __attribute__((ext_vector_type(2)));
typedef float v8f __attribute__((ext_vector_type(8)));

#define B_   8
#define C_   64
#define H_   128
#define W_   128
#define P_   (H_ * W_)      // 16384 pixels per image
#define OC_  16
#define HO_  256
#define WO_  256

// ---------------------------------------------------------------------------
// Kernel 1: offset 1x1 conv (M=pixels, N=32, K=64) on the fp32 WMMA pipe,
// fused with offset->sample-coordinate math (ix = w + off_x, iy = h + off_y,
// border clamp). Writes (ix,iy) float2 per (pixel, k) into workspace,
// pixel-major so k runs contiguous across lanes (coalesced 128B chunks).
// Each wave: 16 pixels x 32 offset channels, two 16x16 accumulators.
// ---------------------------------------------------------------------------
__global__ __launch_bounds__(256) void dysample_coords_kernel(
    const float* __restrict__ x,      // (B, C, H, W)
    const float* __restrict__ ow,     // (32, C)
    const float* __restrict__ ob,     // (32)
    float2* __restrict__ coords)      // (B, P, 16) of (ix, iy)
{
    const int wave  = (blockIdx.x * blockDim.x + threadIdx.x) >> 5;
    const int lane  = threadIdx.x & 31;
    const int m     = lane & 15;               // M (pixel) / N (channel) index
    const int khalf = (lane < 16) ? 0 : 2;     // K sub-offset per A/B layout

    const int b  = wave >> 10;                 // 1024 16-pixel tiles per image
    const int p0 = (wave & 1023) << 4;

    v8f acc0 = {};                             // offset channels 0..15  (x)
    v8f acc1 = {};                             // offset channels 16..31 (y)

    const float* xb  = x + (size_t)b * C_ * P_ + p0 + m;
    const float* ow0 = ow + m * C_;            // row o = m
    const float* ow1 = ow + (m + 16) * C_;     // row o = m + 16

    #pragma unroll
    for (int kk = 0; kk < 16; ++kk) {
        const int kb = kk * 4 + khalf;
        v2f a, b0, b1;
        a.x  = xb[(size_t)kb * P_];            // A[m, kb]   (strided over channels)
        a.y  = xb[(size_t)(kb + 1) * P_];      // A[m, kb+1]
        b0.x = ow0[kb];     b0.y = ow0[kb + 1];
        b1.x = ow1[kb];     b1.y = ow1[kb + 1];
        acc0 = __builtin_amdgcn_wmma_f32_16x16x4_f32(
                   false, a, false, b0, (short)0, acc0, false, false);
        acc1 = __builtin_amdgcn_wmma_f32_16x16x4_f32(
                   false, a, false, b1, (short)0, acc1, false, false);
    }

    // Epilogue: lane holds N = k = g*4 + i*2 + j; VGPR v holds pixel p0+v(+8).
    const int   k   = m;
    const int   gi  = (k >> 1) & 1;            // i (row parity in 2x upsample)
    const int   gj  =  k       & 1;            // j (col parity)
    const float ipx = gj * 0.5f - 0.25f;       // init_pos h[j]
    const float ipy = gi * 0.5f - 0.25f;       // init_pos h[i]
    const float bx  = ob[k];
    const float by  = ob[16 + k];
    const int   prow = (lane < 16) ? 0 : 8;

    float2* cb = coords + (size_t)b * P_ * 16;
    #pragma unroll
    for (int v = 0; v < 8; ++v) {
        const int   p  = p0 + v + prow;        // 16 pixels stay in one row (W=128)
        const float wf = (float)(p & (W_ - 1));
        const float hf = (float)(p >> 7);
        float ix = wf + (acc0[v] + bx) * 0.25f + ipx;
        float iy = hf + (acc1[v] + by) * 0.25f + ipy;
        ix = fminf(fmaxf(ix, 0.0f), (float)(W_ - 1));   // border padding
        iy = fminf(fmaxf(iy, 0.0f), (float)(H_ - 1));
        cb[(size_t)p * 16 + k] = make_float2(ix, iy);
    }
}

// ---------------------------------------------------------------------------
// Kernel 2: bilinear gather + end 1x1 conv (M=16 oc, N=16 out pixels, K=64)
// on the fp32 WMMA pipe. Each wave: 16 consecutive output pixels of one
// output row; each lane owns one output pixel and gathers its 2 channels
// per k-step (B-fragment). A = end_w. D: VGPR v <-> oc, lanes <-> pixels,
// so stores are coalesced per VGPR.
// ---------------------------------------------------------------------------
__global__ __launch_bounds__(256) void dysample_sample_end_kernel(
    const float* __restrict__ x,       // (B, C, H, W)
    const float2* __restrict__ coords, // (B, P, 16)
    const float* __restrict__ ew,      // (OC, C)
    const float* __restrict__ eb,      // (OC)
    float* __restrict__ out)           // (B, OC, HO, WO)
{
    const int wave  = (blockIdx.x * blockDim.x + threadIdx.x) >> 5;
    const int lane  = threadIdx.x & 31;
    const int n     = lane & 15;               // N: this lane's output pixel
    const int khalf = (lane < 16) ? 0 : 2;

    const int b   = wave >> 12;                // 4096 tiles per image (256*256/16)
    const int rem = wave & 4095;
    const int ho  = rem >> 4;
    const int wo  = ((rem & 15) << 4) + n;
    const int i = ho & 1, h = ho >> 1;
    const int j = wo & 1, w = wo >> 1;

    // Per-group bilinear corner indices + weights (coords are pre-clamped).
    int   c00[4], c01[4], c10[4], c11[4];
    float w00[4], w01[4], w10[4], w11[4];
    const float2* cp = coords + ((size_t)b * P_ + h * W_ + w) * 16;
    #pragma unroll
    for (int g = 0; g < 4; ++g) {
        const float2 c  = cp[g * 4 + i * 2 + j];
        const int    x0 = (int)c.x;
        const int    y0 = (int)c.y;
        const int    x1 = min(x0 + 1, W_ - 1);
        const int    y1 = min(y0 + 1, H_ - 1);
        const float  fx = c.x - (float)x0;
        const float  fy = c.y - (float)y0;
        c00[g] = y0 * W_ + x0;   c01[g] = y0 * W_ + x1;
        c10[g] = y1 * W_ + x0;   c11[g] = y1 * W_ + x1;
        w00[g] = (1.0f - fx) * (1.0f - fy);
        w01[g] = fx * (1.0f - fy);
        w10[g] = (1.0f - fx) * fy;
        w11[g] = fx * fy;
    }

    v8f acc = {};
    const float* xb = x + (size_t)b * C_ * P_;
    const float* ewn = ew + n * C_;            // A row: oc = n
    #pragma unroll
    for (int kk = 0; kk < 16; ++kk) {
        const int kb = kk * 4 + khalf;         // channels kb, kb+1 (same group)
        const int g  = kk >> 2;
        const float* ch0 = xb + (size_t)kb * P_;
        const float* ch1 = ch0 + P_;
        v2f bfrag, afrag;
        bfrag.x = w00[g] * ch0[c00[g]] + w01[g] * ch0[c01[g]]
                + w10[g] * ch0[c10[g]] + w11[g] * ch0[c11[g]];
        bfrag.y = w00[g] * ch1[c00[g]] + w01[g] * ch1[c01[g]]
                + w10[g] * ch1[c10[g]] + w11[g] * ch1[c11[g]];
        afrag.x = ewn[kb];
        afrag.y = ewn[kb + 1];
        acc = __builtin_amdgcn_wmma_f32_16x16x4_f32(
                  false, afrag, false, bfrag, (short)0, acc, false, false);
    }

    // D: VGPR v -> oc = v (+8 for high lanes); lanes -> consecutive pixels.
    const int ocbase = (lane < 16) ? 0 : 8;
    #pragma unroll
    for (int v = 0; v < 8; ++v) {
        const int oc = ocbase + v;
        out[((size_t)b * OC_ + oc) * (HO_ * WO_) + (size_t)ho * WO_ + wo]
            = acc[v] + eb[oc];
    }
}

extern "C" void kernel_launch(void* const* d_in, const int* in_sizes, int n_in,
                              void* d_out, int out_size, void* d_ws, size_t ws_size,
                              hipStream_t stream) {
    const float* x  = (const float*)d_in[0];   // (8,64,128,128)
    const float* ow = (const float*)d_in[1];   // (32,64)
    const float* ob = (const float*)d_in[2];   // (32)
    const float* ew = (const float*)d_in[3];   // (16,64)
    const float* eb = (const float*)d_in[4];   // (16)
    float* out      = (float*)d_out;           // (8,16,256,256)
    float2* coords  = (float2*)d_ws;           // needs 8*16384*16*8B = 16 MB

    // Kernel 1: B*P/16 = 8192 waves -> 1024 blocks of 256 (8 waves/block)
    dysample_coords_kernel<<<1024, 256, 0, stream>>>(x, ow, ob, coords);
    // Kernel 2: B*HO*WO/16 = 32768 waves -> 4096 blocks of 256
    dysample_sample_end_kernel<<<4096, 256, 0, stream>>>(x, coords, ew, eb, out);
}